// FuzzyUserAllocator_1_24472723653401
// MI455X (gfx1250) — compile-verified
//
#include <hip/hip_runtime.h>
#include <hip/hip_bf16.h>

#define EMBED 128
#define HEADS 8
#define DH 16
#define NUSERS 4096
#define NSERV 512
#define PADW 132  // 128 + 4: bank-conflict-free for stride-row reads, 8B-aligned rows

typedef float f32x2 __attribute__((ext_vector_type(2)));
typedef float f32x8 __attribute__((ext_vector_type(8)));

static __device__ inline f32x8 wmma4(f32x2 a, f32x2 b, f32x8 c) {
  // D = A(16x4,f32) * B(4x16,f32) + C(16x16,f32)
  return __builtin_amdgcn_wmma_f32_16x16x4_f32(false, a, false, b, (short)0, c, false, false);
}

static __device__ inline float rmax16(float v) {
#pragma unroll
  for (int m = 8; m >= 1; m >>= 1) v = fmaxf(v, __shfl_xor(v, m, 32));
  return v;
}
static __device__ inline float rsum16(float v) {
#pragma unroll
  for (int m = 8; m >= 1; m >>= 1) v += __shfl_xor(v, m, 32);
  return v;
}

// h = x(n x 4, row stride ldx) @ W_in(4x128) + b_in   — single K=4 WMMA per 16x16 tile
__global__ void h_proj(const float* __restrict__ x, int ldx,
                       const float* __restrict__ Win, const float* __restrict__ bin,
                       float* __restrict__ h) {
  const int lane = threadIdx.x;
  const int hf = lane >> 4;      // half-wave: selects K pair {0,1} vs {2,3}
  const int r  = lane & 15;
  const int row0 = blockIdx.x * 16;
  const float* xr = x + (size_t)(row0 + r) * ldx;
  f32x2 a; a.x = xr[2 * hf]; a.y = xr[2 * hf + 1];
#pragma unroll
  for (int jt = 0; jt < 8; ++jt) {
    const int col = jt * 16 + r;
    f32x2 b; b.x = Win[(2 * hf) * EMBED + col]; b.y = Win[(2 * hf + 1) * EMBED + col];
    const float bv = bin[col];
    f32x8 c;
#pragma unroll
    for (int i = 0; i < 8; ++i) c[i] = bv;
    c = wmma4(a, b, c);
#pragma unroll
    for (int i = 0; i < 8; ++i)
      h[(size_t)(row0 + i + 8 * hf) * EMBED + col] = c[i];
  }
}

// out = A(n x 128) @ W(128x128) + bias   — 32 chained K=4 WMMAs per 16x16 tile
__global__ void gemm128_bias(const float* __restrict__ A, const float* __restrict__ W,
                             const float* __restrict__ bias, float* __restrict__ out) {
  const int lane = threadIdx.x;
  const int hf = lane >> 4;
  const int r  = lane & 15;
  const int row0 = blockIdx.x * 16;
  const float* Ar = A + (size_t)(row0 + r) * EMBED;
  f32x2 areg[32];
#pragma unroll
  for (int kk = 0; kk < 32; ++kk) {
    const int kb = 4 * kk + 2 * hf;
    areg[kk].x = Ar[kb]; areg[kk].y = Ar[kb + 1];
  }
  for (int jt = 0; jt < 8; ++jt) {
    const int col = jt * 16 + r;
    const float bv = bias[col];
    f32x8 c;
#pragma unroll
    for (int i = 0; i < 8; ++i) c[i] = bv;
#pragma unroll
    for (int kk = 0; kk < 32; ++kk) {
      const int kb = 4 * kk + 2 * hf;
      f32x2 b; b.x = W[(size_t)kb * EMBED + col]; b.y = W[(size_t)(kb + 1) * EMBED + col];
      c = wmma4(areg[kk], b, c);
    }
#pragma unroll
    for (int i = 0; i < 8; ++i)
      out[(size_t)(row0 + i + 8 * hf) * EMBED + col] = c[i];
  }
}

// Flash attention. Block = 256 threads = 8 waves (one per head), one 16-row stripe.
// K/V tiles staged cooperatively into LDS once per block (8x traffic reduction),
// double-buffered so tile t+1 global loads overlap tile t compute.
__global__ void attn_head(const float* __restrict__ q, const float* __restrict__ k,
                          const float* __restrict__ v, float* __restrict__ o, int n) {
  __shared__ float kbuf[2][16][PADW];
  __shared__ float vbuf[2][16][PADW];
  __shared__ float pbuf[HEADS][16][17];  // per-wave D->A layout bounce for P
  const int tid = threadIdx.x;
  const int hd = tid >> 5;
  const int lane = tid & 31;
  const int hf = lane >> 4;
  const int r  = lane & 15;
  const int row0 = blockIdx.x * 16;
  const int d0 = hd * DH;

  f32x2 qa[4];
#pragma unroll
  for (int kk = 0; kk < 4; ++kk) {
    const int kb = 4 * kk + 2 * hf;
    qa[kk].x = q[(size_t)(row0 + r) * EMBED + d0 + kb];
    qa[kk].y = q[(size_t)(row0 + r) * EMBED + d0 + kb + 1];
  }

  f32x8 oacc;
  float mrow[8], lrow[8];
#pragma unroll
  for (int i = 0; i < 8; ++i) { oacc[i] = 0.f; mrow[i] = -__builtin_inff(); lrow[i] = 0.f; }

  const int ntiles = n / 16;
  f32x2 kreg[4], vreg[4];

  // preload tile 0 into buffer 0
#pragma unroll
  for (int j = 0; j < 4; ++j) {
    const int idx = (tid + j * 256) * 2;   // 1024 f32x2 chunks of the 16x128 tile
    kreg[j] = *(const f32x2*)(k + idx);
    vreg[j] = *(const f32x2*)(v + idx);
  }
#pragma unroll
  for (int j = 0; j < 4; ++j) {
    const int idx = (tid + j * 256) * 2;
    const int row = idx >> 7, col = idx & 127;
    *(f32x2*)(&kbuf[0][row][col]) = kreg[j];
    *(f32x2*)(&vbuf[0][row][col]) = vreg[j];
  }
  __syncthreads();

  for (int ct = 0; ct < ntiles; ++ct) {
    const int cur = ct & 1;
    const bool havenext = (ct + 1) < ntiles;
    if (havenext) {
      const size_t nb = (size_t)(ct + 1) * 16 * EMBED;
#pragma unroll
      for (int j = 0; j < 4; ++j) {
        const int idx = (tid + j * 256) * 2;
        kreg[j] = *(const f32x2*)(k + nb + idx);
        vreg[j] = *(const f32x2*)(v + nb + idx);
      }
    }
    // ---- S = q k^T (16x16 tile) ----
    f32x8 s;
#pragma unroll
    for (int i = 0; i < 8; ++i) s[i] = 0.f;
#pragma unroll
    for (int kk = 0; kk < 4; ++kk) {
      const int kb = 4 * kk + 2 * hf;
      f32x2 b; b.x = kbuf[cur][r][d0 + kb]; b.y = kbuf[cur][r][d0 + kb + 1];
      s = wmma4(qa[kk], b, s);
    }
    // ---- online softmax per row (row i+8*hf lives in this 16-lane group) ----
#pragma unroll
    for (int i = 0; i < 8; ++i) {
      const float sv = s[i] * 0.25f;
      const float tmax = rmax16(sv);
      const float mnew = fmaxf(mrow[i], tmax);
      const float p = expf(sv - mnew);
      const float scale = expf(mrow[i] - mnew);
      const float tsum = rsum16(p);
      lrow[i] = lrow[i] * scale + tsum;
      oacc[i] = oacc[i] * scale;
      mrow[i] = mnew;
      s[i] = p;
    }
    // ---- P: D-layout -> A-layout via wave-private LDS (DS in-order per wave) ----
#pragma unroll
    for (int i = 0; i < 8; ++i) pbuf[hd][i + 8 * hf][r] = s[i];
    asm volatile("" ::: "memory");
#pragma unroll
    for (int kk = 0; kk < 4; ++kk) {
      const int kb = 4 * kk + 2 * hf;
      f32x2 pa; pa.x = pbuf[hd][r][kb]; pa.y = pbuf[hd][r][kb + 1];
      f32x2 b; b.x = vbuf[cur][kb][d0 + r]; b.y = vbuf[cur][kb + 1][d0 + r];
      oacc = wmma4(pa, b, oacc);
    }
    if (havenext) {
#pragma unroll
      for (int j = 0; j < 4; ++j) {
        const int idx = (tid + j * 256) * 2;
        const int row = idx >> 7, col = idx & 127;
        *(f32x2*)(&kbuf[cur ^ 1][row][col]) = kreg[j];
        *(f32x2*)(&vbuf[cur ^ 1][row][col]) = vreg[j];
      }
    }
    __syncthreads();
  }
#pragma unroll
  for (int i = 0; i < 8; ++i)
    o[(size_t)(row0 + i + 8 * hf) * EMBED + d0 + r] = oacc[i] / lrow[i];
}

__global__ void row_mean(const float* __restrict__ ctx, float* __restrict__ sc, int n) {
  const int i = blockIdx.x * blockDim.x + threadIdx.x;
  if (i < n) {
    float s = 0.f;
#pragma unroll 8
    for (int j = 0; j < EMBED; ++j) s += ctx[(size_t)i * EMBED + j];
    sc[i] = s * (1.0f / EMBED);
  }
}

// Stable rank = argsort(-score): ties broken by ascending index
__global__ void rank_users(const float* __restrict__ sc, int* __restrict__ order, int n) {
  const int u = blockIdx.x * blockDim.x + threadIdx.x;
  if (u < n) {
    const float su = sc[u];
    int rank = 0;
    for (int j = 0; j < n; ++j) {
      const float sj = sc[j];
      rank += (sj > su) || (sj == su && j < u);
    }
    order[rank] = u;
  }
}

static __device__ inline float bce_logits(float x, float y) {
  return fmaxf(x, 0.f) + log1pf(expf(-fabsf(x))) - x * y;
}

// One wave: sequential greedy allocation with LDS-resident capacities.
__global__ void alloc_scan(const float* __restrict__ servers, const float* __restrict__ users,
                           const int* __restrict__ masks, const int* __restrict__ order,
                           const float* __restrict__ sscores, float* __restrict__ out) {
  __shared__ float cap[NSERV][4];
  __shared__ float usage[NSERV];
  __shared__ float sscl[NSERV];
  const int lane = threadIdx.x;
  for (int s = lane; s < NSERV; s += 32) {
#pragma unroll
    for (int j = 0; j < 4; ++j) cap[s][j] = servers[s * 7 + 3 + j];
    usage[s] = 0.f;
    sscl[s] = sscores[s];
  }
  __syncthreads();
  float* allocOut = out + 1;
  int allocCount = 0;
  for (int t = 0; t < NUSERS; ++t) {
    const int u = order[t];
    const float w0 = users[u * 6 + 2], w1 = users[u * 6 + 3];
    const float w2 = users[u * 6 + 4], w3 = users[u * 6 + 5];
    float best = -__builtin_inff();
    int bidx = 0x7fffffff;
    for (int s = lane; s < NSERV; s += 32) {
      const bool feas = (masks[(size_t)u * NSERV + s] != 0) &&
                        cap[s][0] >= w0 && cap[s][1] >= w1 &&
                        cap[s][2] >= w2 && cap[s][3] >= w3;
      if (feas) {
        const float scv = sscl[s];
        if (scv > best || (scv == best && s < bidx)) { best = scv; bidx = s; }
      }
    }
#pragma unroll
    for (int m = 16; m >= 1; m >>= 1) {
      const float ob = __shfl_xor(best, m, 32);
      const int oi = __shfl_xor(bidx, m, 32);
      if (ob > best || (ob == best && oi < bidx)) { best = ob; bidx = oi; }
    }
    const bool ok = best > -1.0f;
    if (lane == 0) {
      allocOut[u] = ok ? (float)bidx : -1.0f;
      if (ok) {
        cap[bidx][0] -= w0; cap[bidx][1] -= w1; cap[bidx][2] -= w2; cap[bidx][3] -= w3;
        usage[bidx] += 1.0f;
        allocCount += 1;
      }
    }
    __syncthreads();
  }
  int used = 0;
  for (int s = lane; s < NSERV; s += 32) used += (usage[s] != 0.f) ? 1 : 0;
#pragma unroll
  for (int m = 16; m >= 1; m >>= 1) used += __shfl_xor(used, m, 32);
  for (int s = lane; s < NSERV; s += 32) out[1 + NUSERS + s] = usage[s];
  if (lane == 0) {
    const float uap = (float)allocCount / (float)NUSERS;
    const float sup = (float)used / (float)NSERV;
    const float total = 2.f * (bce_logits(uap, 1.f) + bce_logits(sup, 0.f));
    // sum(softmax(ctx,0)*total)+sum(softmax(sctx,0)*total) = total*(128+128)
    out[0] = 256.f * total;
    out[1 + NUSERS + NSERV] = uap;
    out[1 + NUSERS + NSERV + 1] = sup;
  }
}

extern "C" void kernel_launch(void* const* d_in, const int* in_sizes, int n_in,
                              void* d_out, int out_size, void* d_ws, size_t ws_size,
                              hipStream_t stream) {
  const float* servers = (const float*)d_in[0];
  const float* users   = (const float*)d_in[1];
  const int*   masks   = (const int*)d_in[2];
  const float* W_in = (const float*)d_in[3];
  const float* b_in = (const float*)d_in[4];
  const float* Wq = (const float*)d_in[5];  const float* bq = (const float*)d_in[6];
  const float* Wk = (const float*)d_in[7];  const float* bk = (const float*)d_in[8];
  const float* Wv = (const float*)d_in[9];  const float* bv = (const float*)d_in[10];
  const float* Wo = (const float*)d_in[11]; const float* bo = (const float*)d_in[12];
  float* out = (float*)d_out;

  float* W = (float*)d_ws;
  const size_t NUF = (size_t)NUSERS * EMBED;
  const size_t NSF = (size_t)NSERV * EMBED;
  float* h_u = W;            float* q_u = W + NUF;  float* k_u = W + 2 * NUF;
  float* v_u = W + 3 * NUF;  float* ao_u = W + 4 * NUF;  float* ctx_u = h_u;
  float* base_s = W + 5 * NUF;
  float* h_s = base_s;           float* q_s = base_s + NSF;  float* k_s = base_s + 2 * NSF;
  float* v_s = base_s + 3 * NSF; float* ao_s = base_s + 4 * NSF;  float* ctx_s = h_s;
  float* su = base_s + 5 * NSF;       // user scores (4096)
  float* ss = su + NUSERS;            // server scores (512)
  int* order = (int*)(ss + NSERV);    // sorted user order (4096)

  // --- user attention ---
  h_proj<<<NUSERS / 16, 32, 0, stream>>>(users + 2, 6, W_in, b_in, h_u);
  gemm128_bias<<<NUSERS / 16, 32, 0, stream>>>(h_u, Wq, bq, q_u);
  gemm128_bias<<<NUSERS / 16, 32, 0, stream>>>(h_u, Wk, bk, k_u);
  gemm128_bias<<<NUSERS / 16, 32, 0, stream>>>(h_u, Wv, bv, v_u);
  attn_head<<<NUSERS / 16, 256, 0, stream>>>(q_u, k_u, v_u, ao_u, NUSERS);
  gemm128_bias<<<NUSERS / 16, 32, 0, stream>>>(ao_u, Wo, bo, ctx_u);
  row_mean<<<NUSERS / 128, 128, 0, stream>>>(ctx_u, su, NUSERS);

  // --- server attention ---
  h_proj<<<NSERV / 16, 32, 0, stream>>>(servers + 3, 7, W_in, b_in, h_s);
  gemm128_bias<<<NSERV / 16, 32, 0, stream>>>(h_s, Wq, bq, q_s);
  gemm128_bias<<<NSERV / 16, 32, 0, stream>>>(h_s, Wk, bk, k_s);
  gemm128_bias<<<NSERV / 16, 32, 0, stream>>>(h_s, Wv, bv, v_s);
  attn_head<<<NSERV / 16, 256, 0, stream>>>(q_s, k_s, v_s, ao_s, NSERV);
  gemm128_bias<<<NSERV / 16, 32, 0, stream>>>(ao_s, Wo, bo, ctx_s);
  row_mean<<<NSERV / 128, 128, 0, stream>>>(ctx_s, ss, NSERV);

  // --- ranking + sequential greedy allocation + final loss ---
  rank_users<<<NUSERS / 256, 256, 0, stream>>>(su, order, NUSERS);
  alloc_scan<<<1, 32, 0, stream>>>(servers, users, masks, order, ss, out);
}